// MultigridInterpolator_75436805587562
// MI455X (gfx1250) — compile-verified
//
#include <hip/hip_runtime.h>
#include <stdint.h>

// Multiresolution hash-grid interpolation (Instant-NGP style) for MI455X.
// - Tables (~40 MB total) are L2-resident (192 MB L2); gathers dominate.
// - Levels 0+1 staged into LDS via gfx1250 async global->LDS b128 copies.
// - Per-point 128-B outputs transposed through LDS so NT stores are fully
//   coalesced (512 B contiguous per wave store) -> no partial-line NT
//   writebacks, and the 256 MB output stream stays out of L2.

#define HG_ENTRIES_MASK 0x7FFFFu
#define HG_P1 19349663u
#define HG_P2 83492791u

typedef __attribute__((ext_vector_type(4))) float v4f;
typedef __attribute__((address_space(3))) char lds_char_t;

struct Tabs { const float2* t[16]; };

__device__ __forceinline__ unsigned lds_off_of(void* p) {
  // generic -> LDS addrspace cast, then ptrtoint gives the 32-bit LDS offset
  return (unsigned)(unsigned long long)(lds_char_t*)p;
}

__device__ __forceinline__ void level_dense(float px, float py, float pz,
                                            float G, int dims,
                                            const float2* __restrict__ t,
                                            float& o0, float& o1) {
  float tx = px * G, ty = py * G, tz = pz * G;
  int ix = (int)tx, iy = (int)ty, iz = (int)tz;
  float fx = tx - (float)ix, fy = ty - (float)iy, fz = tz - (float)iz;
  float gx = 1.0f - fx, gy = 1.0f - fy, gz = 1.0f - fz;
  const int d2 = dims * dims;
  const int base = ix + dims * iy + d2 * iz;
  float a0 = 0.0f, a1 = 0.0f;
#pragma unroll
  for (int c = 0; c < 8; ++c) {
    const int b0 = c & 1, b1 = (c >> 1) & 1, b2 = (c >> 2) & 1;
    const int idx = base + b0 + b1 * dims + b2 * d2;
    float w = (b0 ? fx : gx) * (b1 ? fy : gy) * (b2 ? fz : gz);
    float2 v = t[idx];
    a0 = fmaf(w, v.x, a0);
    a1 = fmaf(w, v.y, a1);
  }
  o0 = a0; o1 = a1;
}

__device__ __forceinline__ void level_hash(float px, float py, float pz,
                                           float G,
                                           const float2* __restrict__ t,
                                           float& o0, float& o1) {
  float tx = px * G, ty = py * G, tz = pz * G;
  int ix = (int)tx, iy = (int)ty, iz = (int)tz;
  float fx = tx - (float)ix, fy = ty - (float)iy, fz = tz - (float)iz;
  float gx = 1.0f - fx, gy = 1.0f - fy, gz = 1.0f - fz;
  // int64 hash with xor/mod-2^19 reduces exactly to 32-bit arithmetic:
  // only the low 19 bits of each term survive the xor + mask.
  unsigned a0 = (unsigned)ix;
  unsigned a1 = (unsigned)iy * HG_P1, a1p = a1 + HG_P1;
  unsigned a2 = (unsigned)iz * HG_P2, a2p = a2 + HG_P2;
  float s0 = 0.0f, s1 = 0.0f;
#pragma unroll
  for (int c = 0; c < 8; ++c) {
    const int b0 = c & 1, b1 = (c >> 1) & 1, b2 = (c >> 2) & 1;
    unsigned h = ((a0 + (unsigned)b0) ^ (b1 ? a1p : a1) ^ (b2 ? a2p : a2)) &
                 HG_ENTRIES_MASK;
    float w = (b0 ? fx : gx) * (b1 ? fy : gy) * (b2 ? fz : gz);
    float2 v = t[h];
    s0 = fmaf(w, v.x, s0);
    s1 = fmaf(w, v.y, s1);
  }
  o0 = s0; o1 = s1;
}

__global__ __launch_bounds__(256, 2)
void hashgrid_kernel(const float* __restrict__ x, float* __restrict__ out,
                     Tabs tabs, int npts) {
  // level0 table: 17^3*8 = 39304 B -> padded to 40960 (10 * 4096)
  // level1 table: 21^3*8 = 74088 B -> padded to 77824 (19 * 4096)
  __shared__ float2 sh0[5120];      // 40960 B
  __shared__ float2 sh1[9728];      // 77824 B
  __shared__ float  shout[256 * 33]; // 33792 B transpose buffer (stride 33:
                                     // conflict-free write AND read phases)

  const unsigned tid = threadIdx.x;

  // ---- CDNA5 async global->LDS staging (ASYNCcnt path) ----
  {
    uint64_t g0 = (uint64_t)(uintptr_t)tabs.t[0];
    uint64_t g1 = (uint64_t)(uintptr_t)tabs.t[1];
    unsigned dst = lds_off_of((void*)sh0) + tid * 16u;
    unsigned off = tid * 16u;
#pragma unroll
    for (int i = 0; i < 10; ++i) {
      asm volatile("global_load_async_to_lds_b128 %0, %1, %2 offset:0"
                   :: "v"(dst), "v"(off), "s"(g0) : "memory");
      dst += 4096u; off += 4096u;
    }
    dst = lds_off_of((void*)sh1) + tid * 16u;
    off = tid * 16u;
#pragma unroll
    for (int i = 0; i < 19; ++i) {
      asm volatile("global_load_async_to_lds_b128 %0, %1, %2 offset:0"
                   :: "v"(dst), "v"(off), "s"(g1) : "memory");
      dst += 4096u; off += 4096u;
    }
    asm volatile("s_wait_asynccnt 0" ::: "memory");
  }
  __syncthreads();

  const unsigned total = gridDim.x * blockDim.x;
  const unsigned iters = ((unsigned)npts + total - 1u) / total;

  for (unsigned it = 0; it < iters; ++it) {
    const unsigned nb = it * total + blockIdx.x * 256u;  // block's base point
    const unsigned n  = nb + tid;
    const bool active = n < (unsigned)npts;

    float o[32];
    if (active) {
      // prefetch next iteration's coords (speculative, safe if OOB)
      __builtin_prefetch(x + 3ull * (n + total), 0, 1);

      const float px = __builtin_nontemporal_load(x + 3ull * n + 0);
      const float py = __builtin_nontemporal_load(x + 3ull * n + 1);
      const float pz = __builtin_nontemporal_load(x + 3ull * n + 2);

      // dense levels (grid sizes 16,20,25,32,40,51,64; dims = g+1)
      level_dense(px, py, pz,  16.0f, 17, (const float2*)sh0, o[0],  o[1]);
      level_dense(px, py, pz,  20.0f, 21, (const float2*)sh1, o[2],  o[3]);
      level_dense(px, py, pz,  25.0f, 26, tabs.t[2],          o[4],  o[5]);
      level_dense(px, py, pz,  32.0f, 33, tabs.t[3],          o[6],  o[7]);
      level_dense(px, py, pz,  40.0f, 41, tabs.t[4],          o[8],  o[9]);
      level_dense(px, py, pz,  51.0f, 52, tabs.t[5],          o[10], o[11]);
      level_dense(px, py, pz,  64.0f, 65, tabs.t[6],          o[12], o[13]);
      // hashed levels (grid sizes 81,102,128,161,203,256,323,406,512)
      level_hash (px, py, pz,  81.0f, tabs.t[7],  o[14], o[15]);
      level_hash (px, py, pz, 102.0f, tabs.t[8],  o[16], o[17]);
      level_hash (px, py, pz, 128.0f, tabs.t[9],  o[18], o[19]);
      level_hash (px, py, pz, 161.0f, tabs.t[10], o[20], o[21]);
      level_hash (px, py, pz, 203.0f, tabs.t[11], o[22], o[23]);
      level_hash (px, py, pz, 256.0f, tabs.t[12], o[24], o[25]);
      level_hash (px, py, pz, 323.0f, tabs.t[13], o[26], o[27]);
      level_hash (px, py, pz, 406.0f, tabs.t[14], o[28], o[29]);
      level_hash (px, py, pz, 512.0f, tabs.t[15], o[30], o[31]);
    }

    __syncthreads();  // previous iteration's shout reads are done
    if (active) {
      float* row = shout + tid * 33u;
#pragma unroll
      for (int i = 0; i < 32; ++i) row[i] = o[i];
    }
    __syncthreads();  // shout fully written

    // Coalesced NT store: block writes 256 points * 128 B = 32 KB contiguous.
    // Lane handles float index m = i*1024 + tid*4 of the block region; a wave
    // store instruction covers 512 B contiguous -> full cachelines, NT-safe.
    float* ob = out + (size_t)nb * 32u;
#pragma unroll
    for (int i = 0; i < 8; ++i) {
      const unsigned m = (unsigned)i * 1024u + tid * 4u;
      const unsigned p = m >> 5;        // local point
      const unsigned c = m & 31u;       // component (multiple of 4)
      if (nb + p < (unsigned)npts) {
        const float* s = shout + p * 33u + c;
        v4f v = {s[0], s[1], s[2], s[3]};
        __builtin_nontemporal_store(v, (v4f*)(ob + m));
      }
    }
  }
}

extern "C" void kernel_launch(void* const* d_in, const int* in_sizes, int n_in,
                              void* d_out, int out_size, void* d_ws,
                              size_t ws_size, hipStream_t stream) {
  (void)n_in; (void)out_size; (void)d_ws; (void)ws_size;
  Tabs tabs;
  for (int i = 0; i < 16; ++i) tabs.t[i] = (const float2*)d_in[1 + i];
  const float* x = (const float*)d_in[0];
  const int npts = in_sizes[0] / 3;
  hipLaunchKernelGGL(hashgrid_kernel, dim3(2048), dim3(256), 0, stream,
                     x, (float*)d_out, tabs, npts);
}